// RelationalGraphletConvolution_30477087932481
// MI455X (gfx1250) — compile-verified
//
#include <hip/hip_runtime.h>

typedef __attribute__((ext_vector_type(2))) float v2f;
typedef __attribute__((ext_vector_type(4))) float v4f;
typedef __attribute__((ext_vector_type(8))) float v8f;

#define NB 64
#define NN 32
#define NR 16
#define NG 4960          // C(32,3)
#define NF 16
#define NTILES 310       // NG / 16
#define TILES_PER_BLOCK 40
#define TILES_PER_WAVE 5

__global__ __launch_bounds__(256)
void graphlet_conv_wmma(const float* __restrict__ inp,
                        const float* __restrict__ flt,
                        float* __restrict__ out) {
  // inputs[b] slice: 32*32*16 f32 = 64 KB; filters: 16*9*16 f32 = 9 KB.
  __shared__ __align__(16) float s_in[NN * NN * NR];
  __shared__ __align__(16) float s_flt[NF * 9 * NR];

  const int tid  = threadIdx.x;
  const int bIdx = blockIdx.y;
  const int wave = tid >> 5;
  const int lane = tid & 31;
  const int half = lane >> 4;   // which K-pair half of the wave
  const int col  = lane & 15;   // A row (group) / B col (filter)

  // ---- Stage inputs[bIdx] and filters into LDS (float4 cooperative copy) ----
  {
    const v4f* src = (const v4f*)(inp + (size_t)bIdx * (NN * NN * NR));
    v4f* dst = (v4f*)s_in;
#pragma unroll
    for (int i = 0; i < (NN * NN * NR) / 4 / 256; ++i)
      dst[tid + i * 256] = src[tid + i * 256];

    const v4f* fsrc = (const v4f*)flt;
    v4f* fdst = (v4f*)s_flt;
    for (int i = tid; i < (NF * 9 * NR) / 4; i += 256)
      fdst[i] = fsrc[i];
  }
  __syncthreads();

  for (int t = 0; t < TILES_PER_WAVE; ++t) {
    const int tile = blockIdx.x * TILES_PER_BLOCK + wave * TILES_PER_WAVE + t;
    if (tile >= NTILES) break;           // wave-uniform guard: EXEC stays all-1s

    // ---- Unrank combination index g -> (n0 < n1 < n2), lexicographic ----
    const int g = tile * 16 + col;
    int rem = g, n0 = 0, n1, n2;
    for (; n0 < 30; ++n0) {
      const int cnt = ((31 - n0) * (30 - n0)) >> 1;   // C(31-n0, 2)
      if (rem < cnt) break;
      rem -= cnt;
    }
    n1 = n0 + 1;
    for (; n1 < 31; ++n1) {
      const int cnt = 31 - n1;
      if (rem < cnt) break;
      rem -= cnt;
    }
    n2 = n1 + 1 + rem;

    // 9 per-lane LDS row bases, one per (i,j); fully unrolled -> registers.
    int rowb[9];
#pragma unroll
    for (int i = 0; i < 3; ++i) {
      const int ni = (i == 0) ? n0 : ((i == 1) ? n1 : n2);
#pragma unroll
      for (int j = 0; j < 3; ++j) {
        const int nj = (j == 0) ? n0 : ((j == 1) ? n1 : n2);
        rowb[i * 3 + j] = (ni * NN + nj) * NR + 2 * half;
      }
    }
    const int fbase = col * (9 * NR) + 2 * half;

    // ---- K = 144 as 36 steps of V_WMMA_F32_16X16X4_F32 ----
    v8f c = {0.f, 0.f, 0.f, 0.f, 0.f, 0.f, 0.f, 0.f};
#pragma unroll
    for (int kk = 0; kk < 36; ++kk) {
      const int ij = kk >> 2;            // uniform (i*3+j) for this step
      const int rb = (kk & 3) << 2;      // r-base within the 16-wide R block
      const v2f a = *(const v2f*)&s_in [rowb[ij] + rb];
      const v2f b = *(const v2f*)&s_flt[fbase + ij * NR + rb];
      c = __builtin_amdgcn_wmma_f32_16x16x4_f32(
              /*neg_a=*/false, a, /*neg_b=*/false, b,
              /*c_mod=*/(short)0, c, /*reuse_a=*/false, /*reuse_b=*/false);
    }

    // ---- Store D: VGPR r -> row M = r + 8*half, col N = col ----
    float* op = out + ((size_t)bIdx * NG + (size_t)tile * 16 + half * 8) * NF + col;
#pragma unroll
    for (int r = 0; r < 8; ++r)
      op[r * NF] = c[r];
  }
}

extern "C" void kernel_launch(void* const* d_in, const int* in_sizes, int n_in,
                              void* d_out, int out_size, void* d_ws, size_t ws_size,
                              hipStream_t stream) {
  const float* inp = (const float*)d_in[0];   // (64, 32, 32, 16) f32
  const float* flt = (const float*)d_in[1];   // (16, 3, 3, 16) f32
  float* out = (float*)d_out;                 // (64, 4960, 16) f32
  dim3 grid(8, NB);                           // x: tile chunk, y: batch
  dim3 block(256);                            // 8 waves of 32
  hipLaunchKernelGGL(graphlet_conv_wmma, grid, block, 0, stream, inp, flt, out);
}